// KCLWONegLoss_54855322304750
// MI455X (gfx1250) — compile-verified
//
#include <hip/hip_runtime.h>
#include <hip/hip_bf16.h>
#include <math.h>

typedef __attribute__((ext_vector_type(2))) float v2f;
typedef __attribute__((ext_vector_type(8))) float v8f;

#define DGRP   64
#define NGRP   256
#define FDIM   512
#define LOFF   8
#define KNEG   32
#define PPAIR  56      // DGRP - LOFF
#define EPSV   1e-8f
#define INVT   10.0f   // 1 / TEMP

// ---------------------------------------------------------------------------
// Kernel 1: sums[d][dim] = sum_ng embs[d][ng][dim]
// Implemented as ones-matmul with V_WMMA_F32_16X16X4_F32, accumulating C over
// 64 chained WMMAs per (group, 16-dim tile). One wave per tile; 2048 tiles.
// ---------------------------------------------------------------------------
__global__ __launch_bounds__(256) void k_sums_wmma(const float* __restrict__ embs,
                                                   float* __restrict__ sums) {
  const int lane  = threadIdx.x & 31;
  const int wave  = threadIdx.x >> 5;
  const int tile  = blockIdx.x * 8 + wave;   // 0..2047
  const int d     = tile >> 5;               // 32 dim-tiles per group
  const int dim0  = (tile & 31) << 4;

  const int col   = lane & 15;               // N within the tile
  const int khalf = lane >> 4;               // which k-row pair this half-wave holds

  const float* base = embs + (size_t)d * NGRP * FDIM + dim0 + col;

  v2f a; a.x = 1.0f; a.y = 1.0f;             // all-ones A matrix (16x4)
  v8f c = {};

  #pragma unroll 4
  for (int ng0 = 0; ng0 < NGRP; ng0 += 4) {
    v2f b;
    b.x = base[(size_t)(ng0 + khalf)     * FDIM];   // k rows {0,1}
    b.y = base[(size_t)(ng0 + 2 + khalf) * FDIM];   // k rows {2,3}
    // D = ones(16x4) * B(4x16) + C  -> every row of D = column sums of B
    c = __builtin_amdgcn_wmma_f32_16x16x4_f32(false, a, false, b,
                                              (short)0, c, false, false);
  }
  // Row M=0 of D: VGPR 0, lanes 0..15 (N = lane).
  if (lane < 16) sums[d * FDIM + dim0 + lane] = c[0];
}

// ---------------------------------------------------------------------------
// Kernel 2: pos[p] = cos(sums[p], sums[p+L]) with torch-style eps clamp.
// One block per pair, 256 threads reduce 512 dims.
// ---------------------------------------------------------------------------
__global__ __launch_bounds__(256) void k_pos(const float* __restrict__ sums,
                                             float* __restrict__ pos) {
  const int p = blockIdx.x;
  const float* si = sums + (size_t)p * FDIM;
  const float* sj = sums + (size_t)(p + LOFF) * FDIM;

  float dot = 0.f, na = 0.f, nb = 0.f;
  for (int d = threadIdx.x; d < FDIM; d += 256) {
    float x = si[d], y = sj[d];
    dot = fmaf(x, y, dot);
    na  = fmaf(x, x, na);
    nb  = fmaf(y, y, nb);
  }
  __shared__ float r0[256], r1[256], r2[256];
  r0[threadIdx.x] = dot; r1[threadIdx.x] = na; r2[threadIdx.x] = nb;
  __syncthreads();
  for (int s = 128; s > 0; s >>= 1) {
    if (threadIdx.x < s) {
      r0[threadIdx.x] += r0[threadIdx.x + s];
      r1[threadIdx.x] += r1[threadIdx.x + s];
      r2[threadIdx.x] += r2[threadIdx.x + s];
    }
    __syncthreads();
  }
  if (threadIdx.x == 0) {
    float n1 = fmaxf(sqrtf(r1[0]), EPSV);
    float n2 = fmaxf(sqrtf(r2[0]), EPSV);
    pos[p] = r0[0] / (n1 * n2);
  }
}

// ---------------------------------------------------------------------------
// Kernel 3: S[p] = sum_d exp(sim_neg[p,d] / T)
// sim_neg[p,d] = cosine over K of g1[neg1[p,:]][:,d] vs g2[neg2[p,:]][:,d].
// One block per pair; thread t owns dims t and t+256 (coalesced gathers).
// ---------------------------------------------------------------------------
__global__ __launch_bounds__(256) void k_neg(const float* __restrict__ g1,
                                             const float* __restrict__ g2,
                                             const int* __restrict__ neg1,
                                             const int* __restrict__ neg2,
                                             float* __restrict__ Sout) {
  const int p = blockIdx.x;
  __shared__ int i1[KNEG], i2[KNEG];
  if (threadIdx.x < KNEG)            i1[threadIdx.x]        = neg1[p * KNEG + threadIdx.x];
  else if (threadIdx.x < 2 * KNEG)   i2[threadIdx.x - KNEG] = neg2[p * KNEG + threadIdx.x - KNEG];
  __syncthreads();

  const int d0 = threadIdx.x;
  const int d1 = threadIdx.x + 256;

  float dot0 = 0.f, na0 = 0.f, nb0 = 0.f;
  float dot1 = 0.f, na1 = 0.f, nb1 = 0.f;

  for (int k = 0; k < KNEG; ++k) {
    const float* ra = g1 + (size_t)i1[k] * FDIM;
    const float* rb = g2 + (size_t)i2[k] * FDIM;
    if (k + 1 < KNEG) {   // global_prefetch_b8 of next gathered rows
      __builtin_prefetch(g1 + (size_t)i1[k + 1] * FDIM + d0);
      __builtin_prefetch(g2 + (size_t)i2[k + 1] * FDIM + d0);
    }
    float a0 = ra[d0], b0 = rb[d0];
    float a1 = ra[d1], b1 = rb[d1];
    dot0 = fmaf(a0, b0, dot0); na0 = fmaf(a0, a0, na0); nb0 = fmaf(b0, b0, nb0);
    dot1 = fmaf(a1, b1, dot1); na1 = fmaf(a1, a1, na1); nb1 = fmaf(b1, b1, nb1);
  }

  float s0 = dot0 / (fmaxf(sqrtf(na0), EPSV) * fmaxf(sqrtf(nb0), EPSV));
  float s1 = dot1 / (fmaxf(sqrtf(na1), EPSV) * fmaxf(sqrtf(nb1), EPSV));
  float acc = expf(s0 * INVT) + expf(s1 * INVT);

  __shared__ float red[256];
  red[threadIdx.x] = acc;
  __syncthreads();
  for (int s = 128; s > 0; s >>= 1) {
    if (threadIdx.x < s) red[threadIdx.x] += red[threadIdx.x + s];
    __syncthreads();
  }
  if (threadIdx.x == 0) Sout[p] = red[0];
}

// ---------------------------------------------------------------------------
// Kernel 4: total = 2 * sum_p [ log(exp(pos_p/T) + S_p) - pos_p/T ]
// ---------------------------------------------------------------------------
__global__ __launch_bounds__(64) void k_final(const float* __restrict__ pos,
                                              const float* __restrict__ S,
                                              float* __restrict__ out) {
  __shared__ float r[64];
  const int t = threadIdx.x;
  float loss = 0.f;
  if (t < PPAIR) {
    float pv = pos[t] * INVT;
    loss = logf(expf(pv) + S[t]) - pv;   // == -log(num/den)
  }
  r[t] = loss;
  __syncthreads();
  for (int s = 32; s > 0; s >>= 1) {
    if (t < s) r[t] += r[t + s];
    __syncthreads();
  }
  if (t == 0) out[0] = 2.0f * r[0];
}

// ---------------------------------------------------------------------------
extern "C" void kernel_launch(void* const* d_in, const int* in_sizes, int n_in,
                              void* d_out, int out_size, void* d_ws, size_t ws_size,
                              hipStream_t stream) {
  const float* embs = (const float*)d_in[0];
  // d_in[1] = g0 is dead code in the reference output path.
  const float* g1   = (const float*)d_in[2];
  const float* g2   = (const float*)d_in[3];
  const int*   neg1 = (const int*)d_in[4];
  const int*   neg2 = (const int*)d_in[5];
  float* out = (float*)d_out;

  float* ws_sums = (float*)d_ws;                 // 64*512 floats
  float* ws_pos  = ws_sums + DGRP * FDIM;        // 56 floats (use 64 slots)
  float* ws_S    = ws_pos + 64;                  // 56 floats

  k_sums_wmma<<<256, 256, 0, stream>>>(embs, ws_sums);
  k_pos<<<PPAIR, 256, 0, stream>>>(ws_sums, ws_pos);
  k_neg<<<PPAIR, 256, 0, stream>>>(g1, g2, neg1, neg2, ws_S);
  k_final<<<1, 64, 0, stream>>>(ws_pos, ws_S, out);
}